// MASGCNBertClassifier_30356828848197
// MI455X (gfx1250) — compile-verified
//
#include <hip/hip_runtime.h>

#define B_ 16
#define L_ 512
#define D_ 768
#define A_ 100
#define H_ 5
#define DK_ 20
#define DEP_ 45
#define NPAD_ 112  // 7 n-tiles of 16, zero-padded

typedef __attribute__((ext_vector_type(2))) float v2f;
typedef __attribute__((ext_vector_type(8))) float v8f;

__device__ __forceinline__ v8f wmma_f32_16x16x4(v2f a, v2f b, v8f c) {
  // 8 args: (neg_a, A, neg_b, B, c_mod, C, reuse_a, reuse_b)
  return __builtin_amdgcn_wmma_f32_16x16x4_f32(false, a, false, b, (short)0, c,
                                               false, false);
}

__device__ __forceinline__ float wave_sum(float v) {
  for (int o = 16; o > 0; o >>= 1) v += __shfl_xor(v, o, 32);
  return v;
}
__device__ __forceinline__ float wave_max(float v) {
  for (int o = 16; o > 0; o >>= 1) v = fmaxf(v, __shfl_xor(v, o, 32));
  return v;
}

// Zero-padded transpose: out[b][n][k] = (n<N) ? in[b][k][n] : 0, n in [0,NPAD_)
__global__ void transpose_pad_kernel(const float* __restrict__ in, int K, int N,
                                     float* __restrict__ out, size_t bsi, size_t bso) {
  int idx = blockIdx.x * blockDim.x + threadIdx.x;
  if (idx >= NPAD_ * K) return;
  int n = idx / K, k = idx % K;
  const float* ip = in + (size_t)blockIdx.y * bsi;
  float* op = out + (size_t)blockIdx.y * bso;
  op[(size_t)n * K + k] = (n < N) ? ip[(size_t)k * N + n] : 0.f;
}

// One wave computes a 16x112 slab of C = A[M,K] @ B[K,N] (+bias, +relu).
// B is pre-transposed+padded: Bt[n][k], n in [0,112). LDBT is compile-time so the
// 7 n-tile offsets fold into the global_load_b64 immediate offset (no addr VALU).
// Software-pipelined: loads for step k+1 issue before the 7 WMMAs of step k.
// f32 WMMA 16x16x4 layout (ISA 7.12.2):
//   A: lanes 0-15 m=lane (v0:k0,v1:k0+1); lanes 16-31 m=lane-16 (v0:k0+2,v1:k0+3)
//   B: mirrored with n in place of m;  C/D: vgpr r -> row r+8*(lane>>4), col lane&15
// Batching: A/C indexed by blockIdx.y, B by blockIdx.y/bdiv.
template <int LDBT>
__global__ __launch_bounds__(32) void gemm_n7_kernel(
    const float* __restrict__ A, int lda, size_t abs_,
    const float* __restrict__ Bt, size_t bbs, int bdiv,
    const float* __restrict__ bias, float* __restrict__ C, int ldc, size_t cbs,
    int N, int K, int relu) {
  int lane = threadIdx.x;
  int half = lane >> 4, lanem = lane & 15;
  int m0 = blockIdx.x * 16;
  int bz = blockIdx.y;
  const float* Ab = A + (size_t)bz * abs_;
  const float* Bb = Bt + (size_t)(bz / bdiv) * bbs;
  float* Cb = C + (size_t)bz * cbs;
  v8f zero = {};
  v8f acc[7];
#pragma unroll
  for (int nt = 0; nt < 7; ++nt) acc[nt] = zero;
  const float* ap = Ab + (size_t)(m0 + lanem) * lda + 2 * half;
  const float* bp = Bb + (size_t)lanem * LDBT + 2 * half;
  v2f a0, b0[7];
  a0.x = ap[0];
  a0.y = ap[1];
#pragma unroll
  for (int nt = 0; nt < 7; ++nt) {
    b0[nt].x = bp[nt * 16 * LDBT];
    b0[nt].y = bp[nt * 16 * LDBT + 1];
  }
#pragma unroll 2
  for (int k = 4; k < K; k += 4) {
    v2f a1, b1[7];
    a1.x = ap[k];
    a1.y = ap[k + 1];
#pragma unroll
    for (int nt = 0; nt < 7; ++nt) {
      b1[nt].x = bp[nt * 16 * LDBT + k];
      b1[nt].y = bp[nt * 16 * LDBT + k + 1];
    }
#pragma unroll
    for (int nt = 0; nt < 7; ++nt) acc[nt] = wmma_f32_16x16x4(a0, b0[nt], acc[nt]);
    a0 = a1;
#pragma unroll
    for (int nt = 0; nt < 7; ++nt) b0[nt] = b1[nt];
  }
#pragma unroll
  for (int nt = 0; nt < 7; ++nt) acc[nt] = wmma_f32_16x16x4(a0, b0[nt], acc[nt]);
#pragma unroll
  for (int nt = 0; nt < 7; ++nt) {
    int n = nt * 16 + lanem;
    if (n < N) {
      float bv = bias ? bias[n] : 0.f;
#pragma unroll
      for (int r = 0; r < 8; ++r) {
        int m = m0 + r + 8 * half;
        float v = acc[nt][r] + bv;
        if (relu) v = fmaxf(v, 0.f);
        Cb[(size_t)m * ldc + n] = v;
      }
    }
  }
}

// LayerNorm (unbiased var, std+eps) -> ln_x rows
__global__ __launch_bounds__(256) void ln_kernel(const float* __restrict__ x,
                                                 const float* __restrict__ g,
                                                 const float* __restrict__ bb,
                                                 float* __restrict__ out) {
  __shared__ float red[8];
  int row = blockIdx.x, t = threadIdx.x;
  const float* xr = x + (size_t)row * D_;
  float s = 0.f;
  for (int d = t; d < D_; d += 256) s += xr[d];
  s = wave_sum(s);
  if ((t & 31) == 0) red[t >> 5] = s;
  __syncthreads();
  float mean = 0.f;
  for (int i = 0; i < 8; ++i) mean += red[i];
  mean *= (1.f / D_);
  __syncthreads();
  float v = 0.f;
  for (int d = t; d < D_; d += 256) { float df = xr[d] - mean; v += df * df; }
  v = wave_sum(v);
  if ((t & 31) == 0) red[t >> 5] = v;
  __syncthreads();
  float var = 0.f;
  for (int i = 0; i < 8; ++i) var += red[i];
  var *= (1.f / (D_ - 1));
  float inv = 1.f / (sqrtf(var) + 1e-6f);
  float* orow = out + (size_t)row * D_;
  for (int d = t; d < D_; d += 256) orow[d] = g[d] * (xr[d] - mean) * inv + bb[d];
}

// aspect masked-mean + asp = aspect @ dense_w + dense_b ; also store asp_wn
__global__ __launch_bounds__(128) void aspect_kernel(const float* __restrict__ gcn,
                                                     const float* __restrict__ amask,
                                                     const float* __restrict__ dw,
                                                     const float* __restrict__ db,
                                                     float* __restrict__ asp,
                                                     float* __restrict__ aspwn) {
  int b = blockIdx.x, t = threadIdx.x;
  __shared__ float asp_s[A_];
  __shared__ float wn_s;
  if (t == 0) {
    float wn = 0.f;
    for (int l = 0; l < L_; ++l) wn += amask[b * L_ + l];
    wn_s = wn;
    aspwn[b] = wn;
  }
  __syncthreads();
  for (int a = t; a < A_; a += 128) {
    float s = 0.f;
    for (int l = 0; l < L_; ++l)
      s += gcn[((size_t)b * L_ + l) * A_ + a] * amask[b * L_ + l];
    asp_s[a] = s / wn_s;
  }
  __syncthreads();
  if (t < DK_) {
    float s = db[t];
    for (int a = 0; a < A_; ++a) s += asp_s[a] * dw[a * DK_ + t];
    asp[b * DK_ + t] = s;
  }
}

__global__ void asp_scores_kernel(const float* __restrict__ asp,
                                  const float* __restrict__ k,
                                  const float* __restrict__ bias_m,
                                  float* __restrict__ out) {
  int idx = blockIdx.x * blockDim.x + threadIdx.x;
  if (idx >= B_ * H_ * L_) return;
  int l = idx % L_;
  int h = (idx / L_) % H_;
  int b = idx / (L_ * H_);
  float s = bias_m[0];
  const float* kp = k + ((size_t)b * L_ + l) * A_ + h * DK_;
  const float* ap = asp + b * DK_;
  for (int d = 0; d < DK_; ++d) s += ap[d] * kp[d];
  out[idx] = tanhf(s);
}

__global__ __launch_bounds__(64) void depw_kernel(const float* __restrict__ emb,
                                                  const float* __restrict__ qw,
                                                  const float* __restrict__ qb,
                                                  float* __restrict__ w) {
  __shared__ float s[DEP_];
  int t = threadIdx.x;
  if (t < DEP_) {
    float a = qb[0];
    for (int i = 0; i < A_; ++i) a += emb[t * A_ + i] * qw[i];
    s[t] = a;
  }
  __syncthreads();
  if (t == 0) {
    float mx = -1e30f;
    for (int e = 0; e < DEP_; ++e) mx = fmaxf(mx, s[e]);
    float sum = 0.f;
    for (int e = 0; e < DEP_; ++e) sum += __expf(s[e] - mx);
    float inv = 1.f / sum;
    for (int e = 0; e < DEP_; ++e) w[e] = __expf(s[e] - mx) * inv;
  }
}

// Fused QK^T (WMMA) + scale + asp bias + src mask + short_mask + softmax + syn.
// One wave per (b,h, 16-row tile); 16x512 score tile lives in LDS (32KB).
__global__ __launch_bounds__(32) void scores_kernel(const float* __restrict__ q,
                                                    const float* __restrict__ k,
                                                    const float* __restrict__ asp_sc,
                                                    const int* __restrict__ src_mask,
                                                    const float* __restrict__ short_mask,
                                                    const int* __restrict__ adj,
                                                    const float* __restrict__ depw,
                                                    float* __restrict__ wa) {
  __shared__ float sc[16][L_];
  int bh = blockIdx.y;
  int b = bh / H_;
  int h = bh % H_;
  int i0 = blockIdx.x * 16;
  int lane = threadIdx.x;
  int half = lane >> 4, lanem = lane & 15;
  const float* qp = q + ((size_t)b * L_) * A_ + h * DK_;
  const float* kp = k + ((size_t)b * L_) * A_ + h * DK_ + 2 * half;
  v2f afr[5];
#pragma unroll
  for (int kk = 0; kk < 5; ++kk) {
    int ka = kk * 4 + 2 * half;
    afr[kk].x = qp[(size_t)(i0 + lanem) * A_ + ka];
    afr[kk].y = qp[(size_t)(i0 + lanem) * A_ + ka + 1];
  }
  const float rsc = 0.22360679774997896f;  // 1/sqrt(20)
  for (int jt = 0; jt < 32; ++jt) {
    int j0 = jt * 16;
    const float* kr = kp + (size_t)(j0 + lanem) * A_;
    v2f bfr[5];
#pragma unroll
    for (int kk = 0; kk < 5; ++kk) {  // group loads, wait once per tile
      bfr[kk].x = kr[kk * 4];
      bfr[kk].y = kr[kk * 4 + 1];
    }
    int n = j0 + lanem;
    float aspn = asp_sc[(size_t)bh * L_ + n];
    bool msk = src_mask[b * L_ + n] != 0;
    v8f acc = {};
#pragma unroll
    for (int kk = 0; kk < 5; ++kk) acc = wmma_f32_16x16x4(afr[kk], bfr[kk], acc);
#pragma unroll
    for (int r = 0; r < 8; ++r) {
      int m = r + 8 * half;
      float v = acc[r] * rsc + aspn;
      if (!msk) v = -1e9f;
      v += short_mask[((size_t)b * L_ + (i0 + m)) * L_ + n];
      sc[m][n] = v;
    }
  }
  __syncthreads();
  for (int r = 0; r < 16; ++r) {
    float mx = -1e30f;
    for (int j = lane; j < L_; j += 32) mx = fmaxf(mx, sc[r][j]);
    mx = wave_max(mx);
    float sum = 0.f;
    for (int j = lane; j < L_; j += 32) {
      float e = __expf(sc[r][j] - mx);
      sc[r][j] = e;
      sum += e;
    }
    sum = wave_sum(sum);
    float inv = 1.f / sum;
    int row = i0 + r;
    const int* adjr = adj + ((size_t)b * L_ + row) * L_;
    float* war = wa + ((size_t)bh * L_ + row) * L_;
    for (int j = lane; j < L_; j += 32) {
      int e = adjr[j];
      float syn = (e == 0) ? 0.f : depw[e];
      war[j] = sc[r][j] * inv + syn;
    }
  }
}

__global__ __launch_bounds__(256) void vw_reduce_kernel(const float* __restrict__ Ax,
                                                        float* __restrict__ vwsum) {
  int bh = blockIdx.x, t = threadIdx.x;
  const float* p = Ax + (size_t)bh * L_ * A_;
  float s = 0.f;
  for (int i = t; i < L_ * A_; i += 256) s += p[i];
  s = wave_sum(s);
  __shared__ float red[8];
  if ((t & 31) == 0) red[t >> 5] = s;
  __syncthreads();
  if (t == 0) {
    float tot = 0.f;
    for (int w = 0; w < 8; ++w) tot += red[w];
    vwsum[bh] = tot / (float)(L_ * A_);
  }
}

__global__ __launch_bounds__(32) void vw_mlp_kernel(const float* __restrict__ vwsum,
                                                    const float* __restrict__ f1w,
                                                    const float* __restrict__ f1b,
                                                    const float* __restrict__ f3w,
                                                    const float* __restrict__ f3b,
                                                    float* __restrict__ vw) {
  int b = blockIdx.x;
  if (threadIdx.x != 0) return;
  float in[H_], z[15], o[H_];
  for (int h = 0; h < H_; ++h) in[h] = vwsum[b * H_ + h];
  for (int j = 0; j < 15; ++j) {
    float s = f1b[j];
    for (int h = 0; h < H_; ++h) s += in[h] * f1w[h * 15 + j];
    z[j] = fmaxf(s, 0.f);
  }
  float mx = -1e30f;
  for (int h = 0; h < H_; ++h) {
    float s = f3b[h];
    for (int j = 0; j < 15; ++j) s += z[j] * f3w[j * H_ + h];
    o[h] = s;
    mx = fmaxf(mx, s);
  }
  float sum = 0.f;
  for (int h = 0; h < H_; ++h) sum += __expf(o[h] - mx);
  float inv = 1.f / sum;
  for (int h = 0; h < H_; ++h) vw[b * H_ + h] = __expf(o[h] - mx) * inv;
}

__global__ void combine_kernel(const float* __restrict__ Ax,
                               const float* __restrict__ vw,
                               float* __restrict__ Axc) {
  int idx = blockIdx.x * blockDim.x + threadIdx.x;
  if (idx >= B_ * L_ * A_) return;
  int b = idx / (L_ * A_);
  int la = idx % (L_ * A_);
  float s = 0.f;
  for (int h = 0; h < H_; ++h)
    s += vw[b * H_ + h] * Ax[((size_t)(b * H_ + h)) * L_ * A_ + la];
  Axc[idx] = s;
}

__global__ void g_kernel(const float* __restrict__ gcn, const float* __restrict__ wxw,
                         float* __restrict__ g1, float* __restrict__ g2) {
  int idx = blockIdx.x * blockDim.x + threadIdx.x;
  if (idx >= B_ * L_ * H_) return;
  int h = idx % H_;
  int bl = idx / H_;
  const float* gp = gcn + (size_t)bl * A_;
  float s1 = 0.f, s2 = 0.f;
  for (int a = 0; a < A_; ++a) {
    float g = gp[a];
    s1 += g * wxw[(H_ + a) * H_ + h];
    s2 += g * wxw[(H_ + A_ + a) * H_ + h];
  }
  g1[idx] = s1;
  g2[idx] = s2;
}

// In-place edge update: wa[b,h,i,j] <- sum_h' wa[b,h',i,j]*wxw[h'][h] + g1[b,j,h] + g2[b,i,h] + wxb[h]
__global__ void edge_kernel(float* __restrict__ wa, const float* __restrict__ wxw,
                            const float* __restrict__ wxb,
                            const float* __restrict__ g1,
                            const float* __restrict__ g2) {
  size_t idx = (size_t)blockIdx.x * blockDim.x + threadIdx.x;
  if (idx >= (size_t)B_ * L_ * L_) return;
  int j = (int)(idx % L_);
  size_t bi = idx / L_;
  int i = (int)(bi % L_);
  int b = (int)(bi / L_);
  size_t base = ((size_t)b * H_) * L_ * L_ + (size_t)i * L_ + j;
  float old[H_], nw[H_];
#pragma unroll
  for (int h = 0; h < H_; ++h) old[h] = wa[base + (size_t)h * L_ * L_];
#pragma unroll
  for (int h = 0; h < H_; ++h) {
    float s = wxb[h] + g1[((size_t)b * L_ + j) * H_ + h] + g2[((size_t)b * L_ + i) * H_ + h];
#pragma unroll
    for (int hp = 0; hp < H_; ++hp) s += old[hp] * wxw[hp * H_ + h];
    nw[h] = s;
  }
#pragma unroll
  for (int h = 0; h < H_; ++h) wa[base + (size_t)h * L_ * L_] = nw[h];
}

__global__ __launch_bounds__(128) void final_kernel(const float* __restrict__ gcn,
                                                    const float* __restrict__ amask,
                                                    const float* __restrict__ aspwn,
                                                    const float* __restrict__ cw,
                                                    const float* __restrict__ cb,
                                                    float* __restrict__ out) {
  int b = blockIdx.x, t = threadIdx.x;
  __shared__ float o1[A_];
  for (int a = t; a < A_; a += 128) {
    float s = 0.f;
    for (int l = 0; l < L_; ++l)
      s += fmaxf(gcn[((size_t)b * L_ + l) * A_ + a], 0.f) * amask[b * L_ + l];
    o1[a] = s / aspwn[b];
  }
  __syncthreads();
  if (t < 3) {
    float s = cb[t];
    for (int a = 0; a < A_; ++a) s += o1[a] * cw[a * 3 + t];
    out[b * 3 + t] = s;
  }
}

extern "C" void kernel_launch(void* const* d_in, const int* in_sizes, int n_in,
                              void* d_out, int out_size, void* d_ws, size_t ws_size,
                              hipStream_t stream) {
  (void)in_sizes; (void)n_in; (void)out_size;
  const float* seq        = (const float*)d_in[0];
  const float* short_mask = (const float*)d_in[1];
  const float* amask      = (const float*)d_in[2];
  const int*   smask      = (const int*)d_in[3];
  const int*   adj        = (const int*)d_in[4];
  const float* ln_g       = (const float*)d_in[5];
  const float* ln_b       = (const float*)d_in[6];
  const float* wxx_w      = (const float*)d_in[7];
  const float* wxx_b      = (const float*)d_in[8];
  const float* q_w        = (const float*)d_in[9];
  const float* q_b        = (const float*)d_in[10];
  const float* k_w        = (const float*)d_in[11];
  const float* k_b        = (const float*)d_in[12];
  const float* dense_w    = (const float*)d_in[13];
  const float* dense_b    = (const float*)d_in[14];
  const float* bias_m     = (const float*)d_in[15];
  const float* dep_emb    = (const float*)d_in[16];
  const float* depq_w     = (const float*)d_in[17];
  const float* depq_b     = (const float*)d_in[18];
  const float* W_w        = (const float*)d_in[19];
  const float* W_b        = (const float*)d_in[20];
  const float* wx_w       = (const float*)d_in[21];
  const float* wx_b       = (const float*)d_in[22];
  const float* fc1_w      = (const float*)d_in[23];
  const float* fc1_b      = (const float*)d_in[24];
  const float* fc3_w      = (const float*)d_in[25];
  const float* fc3_b      = (const float*)d_in[26];
  const float* cls_w      = (const float*)d_in[27];
  const float* cls_b      = (const float*)d_in[28];

  char* ws = (char*)d_ws;
  size_t off = 0;
  auto alloc = [&](size_t bytes) -> float* {
    float* p = (float*)(ws + off);
    off = (off + bytes + 255) & ~(size_t)255;
    return p;
  };
  float* ln_x    = alloc((size_t)B_ * L_ * D_ * 4);           // 25.2MB
  float* gcn_in  = alloc((size_t)B_ * L_ * A_ * 4);           // [B,L,A]
  float* qbuf    = alloc((size_t)B_ * L_ * A_ * 4);
  float* kbuf    = alloc((size_t)B_ * L_ * A_ * 4);
  float* asp     = alloc((size_t)B_ * DK_ * 4);
  float* aspwn   = alloc((size_t)B_ * 4);
  float* asp_sc  = alloc((size_t)B_ * H_ * L_ * 4);
  float* depw    = alloc((size_t)DEP_ * 4);
  float* wa      = alloc((size_t)B_ * H_ * L_ * L_ * 4);      // 84MB
  float* Ax      = alloc((size_t)B_ * H_ * L_ * A_ * 4);      // 16.4MB
  float* vwsum   = alloc((size_t)B_ * H_ * 4);
  float* vw      = alloc((size_t)B_ * H_ * 4);
  float* Axc     = alloc((size_t)B_ * L_ * A_ * 4);
  float* gcn_out = alloc((size_t)B_ * L_ * A_ * 4);
  float* g1      = alloc((size_t)B_ * L_ * H_ * 4);
  float* g2      = alloc((size_t)B_ * L_ * H_ * 4);
  float* wxx_t   = alloc((size_t)NPAD_ * D_ * 4);             // transposed+padded weights
  float* qw_t    = alloc((size_t)NPAD_ * A_ * 4);
  float* kw_t    = alloc((size_t)NPAD_ * A_ * 4);
  float* Ww_t    = alloc((size_t)NPAD_ * A_ * 4);
  float* gcn_t   = alloc((size_t)B_ * NPAD_ * L_ * 4);        // 3.7MB
  if (off > ws_size) return;  // workspace too small; bail deterministically

  const int MT = (B_ * L_) / 16;  // 512 row tiles

  // 0) transpose+pad all B operands (one-time, tiny)
  transpose_pad_kernel<<<dim3((NPAD_ * D_ + 255) / 256, 1), 256, 0, stream>>>(wxx_w, D_, A_, wxx_t, 0, 0);
  transpose_pad_kernel<<<dim3((NPAD_ * A_ + 255) / 256, 1), 256, 0, stream>>>(q_w, A_, A_, qw_t, 0, 0);
  transpose_pad_kernel<<<dim3((NPAD_ * A_ + 255) / 256, 1), 256, 0, stream>>>(k_w, A_, A_, kw_t, 0, 0);
  transpose_pad_kernel<<<dim3((NPAD_ * A_ + 255) / 256, 1), 256, 0, stream>>>(W_w, A_, A_, Ww_t, 0, 0);
  // 1) LayerNorm, then WMMA 768x100 projection
  ln_kernel<<<B_ * L_, 256, 0, stream>>>(seq, ln_g, ln_b, ln_x);
  gemm_n7_kernel<D_><<<dim3(MT, 1), 32, 0, stream>>>(ln_x, D_, 0, wxx_t, 0, 1,
                                                     wxx_b, gcn_in, A_, 0, A_, D_, 0);
  // 2) aspect masked-mean + dense
  aspect_kernel<<<B_, 128, 0, stream>>>(gcn_in, amask, dense_w, dense_b, asp, aspwn);
  // 3) Q/K projections (WMMA)
  gemm_n7_kernel<A_><<<dim3(MT, 1), 32, 0, stream>>>(gcn_in, A_, 0, qw_t, 0, 1,
                                                     q_b, qbuf, A_, 0, A_, A_, 0);
  gemm_n7_kernel<A_><<<dim3(MT, 1), 32, 0, stream>>>(gcn_in, A_, 0, kw_t, 0, 1,
                                                     k_b, kbuf, A_, 0, A_, A_, 0);
  // 4) aspect scores + dep weights
  asp_scores_kernel<<<(B_ * H_ * L_ + 255) / 256, 256, 0, stream>>>(asp, kbuf, bias_m, asp_sc);
  depw_kernel<<<1, 64, 0, stream>>>(dep_emb, depq_w, depq_b, depw);
  // 5) fused QK^T + softmax + syn -> weight_adj (WMMA)
  scores_kernel<<<dim3(L_ / 16, B_ * H_), 32, 0, stream>>>(qbuf, kbuf, asp_sc, smask,
                                                           short_mask, adj, depw, wa);
  // 6) two GCN layers
  for (int layer = 0; layer < 2; ++layer) {
    const float* gsrc = (layer == 0) ? gcn_in : gcn_out;
    transpose_pad_kernel<<<dim3((NPAD_ * L_ + 255) / 256, B_), 256, 0, stream>>>(
        gsrc, L_, A_, gcn_t, (size_t)L_ * A_, (size_t)NPAD_ * L_);
    // Ax[b,h] = weight_adj[b,h] (LxL) @ gcn[b] (LxA), batched WMMA
    gemm_n7_kernel<L_><<<dim3(L_ / 16, B_ * H_), 32, 0, stream>>>(
        wa, L_, (size_t)L_ * L_, gcn_t, (size_t)NPAD_ * L_, H_,
        nullptr, Ax, A_, (size_t)L_ * A_, A_, L_, 0);
    vw_reduce_kernel<<<B_ * H_, 256, 0, stream>>>(Ax, vwsum);
    vw_mlp_kernel<<<B_, 32, 0, stream>>>(vwsum, fc1_w, fc1_b, fc3_w, fc3_b, vw);
    combine_kernel<<<(B_ * L_ * A_ + 255) / 256, 256, 0, stream>>>(Ax, vw, Axc);
    gemm_n7_kernel<A_><<<dim3(MT, 1), 32, 0, stream>>>(Axc, A_, 0, Ww_t, 0, 1,
                                                       W_b, gcn_out, A_, 0, A_, A_, 1);
    if (layer + 1 < 2) {  // final edge update is dead code in the reference
      g_kernel<<<(B_ * L_ * H_ + 255) / 256, 256, 0, stream>>>(gcn_out, wx_w, g1, g2);
      edge_kernel<<<(int)(((size_t)B_ * L_ * L_ + 255) / 256), 256, 0, stream>>>(wa, wx_w, wx_b, g1, g2);
    }
  }
  // 7) masked mean + classifier
  final_kernel<<<B_, 128, 0, stream>>>(gcn_out, amask, aspwn, cls_w, cls_b, (float*)d_out);
}